// EmbedMaskPostProcessor_82626580841058
// MI455X (gfx1250) — compile-verified
//
#include <hip/hip_runtime.h>
#include <cstdint>
#include <cstddef>

// ---- problem constants (from reference) ----
#define NIMG 2
#define NC   80            // classes
#define FH   96
#define FW   96
#define HW   (FH*FW)       // 9216
#define EDIM 32            // embedding dim (GEMM K)
#define PIH  192
#define PIW  192
#define OH   384
#define OW   384
#define PP   (OH*OW)       // 147456 pixels
#define KTOP 200
#define KPAD 208           // 13 * 16
#define NFLAT (HW*NC)      // 737280
#define IMGW 768
#define IMGH 768
#define NTILE_X (OW/16)    // 24 pixel-tiles per image row

typedef float v2f __attribute__((ext_vector_type(2)));
typedef float v8f __attribute__((ext_vector_type(8)));

__device__ __forceinline__ float sigm(float x) { return 1.0f / (1.0f + __expf(-x)); }

// ---------------- bilinear 2x upsample: pix [N,32,192,192] -> up [N*32][147456]
__global__ void k_upsample(const float* __restrict__ pix, float* __restrict__ up) {
    int tid = blockIdx.x * blockDim.x + threadIdx.x;
    if (tid >= NIMG * EDIM * PP) return;
    int ch = tid / PP;          // n*32 + d
    int p  = tid % PP;
    int oy = p / OW, ox = p % OW;
    float sx = (float)ox * 0.5f - 0.25f;
    float sy = (float)oy * 0.5f - 0.25f;
    int x0 = (int)floorf(sx); float fx = sx - (float)x0;
    int y0 = (int)floorf(sy); float fy = sy - (float)y0;
    int x0c = min(max(x0, 0), PIW - 1), x1c = min(max(x0 + 1, 0), PIW - 1);
    int y0c = min(max(y0, 0), PIH - 1), y1c = min(max(y0 + 1, 0), PIH - 1);
    const float* b = pix + (size_t)ch * (PIH * PIW);
    float v00 = b[y0c * PIW + x0c], v01 = b[y0c * PIW + x1c];
    float v10 = b[y1c * PIW + x0c], v11 = b[y1c * PIW + x1c];
    up[tid] = (1.0f - fy) * ((1.0f - fx) * v00 + fx * v01)
            +          fy * ((1.0f - fx) * v10 + fx * v11);
}

// ---------------- per-pixel ||p||^2
__global__ void k_psq(const float* __restrict__ up, float* __restrict__ psq) {
    int tid = blockIdx.x * blockDim.x + threadIdx.x;
    if (tid >= NIMG * PP) return;
    int img = tid / PP;
    int p   = tid % PP;
    const float* b = up + (size_t)img * EDIM * PP + p;
    float s = 0.0f;
#pragma unroll
    for (int d = 0; d < EDIM; ++d) { float v = b[(size_t)d * PP]; s += v * v; }
    psq[tid] = s;
}

// ---------------- candidate scores: flat[img][hw*C + c]
__global__ void k_scores(const float* __restrict__ cls, const float* __restrict__ ctr,
                         float* __restrict__ flat) {
    int tid = blockIdx.x * blockDim.x + threadIdx.x;
    if (tid >= NIMG * NFLAT) return;
    int img = tid / NFLAT;
    int rem = tid % NFLAT;
    int hw = rem / NC, c = rem % NC;
    float s  = sigm(cls[((size_t)(img * NC + c)) * HW + hw]);
    float ce = sigm(ctr[img * HW + hw]);
    flat[tid] = (s > 0.05f) ? s * ce : -1.0f;
}

// ---------------- exact sorted top-K + gather (one block per image)
// meta record per k (8 floats): {eSq, margin, valid, pad, b0, b1, b2, b3}
__global__ __launch_bounds__(1024) void k_topk(
    const float* __restrict__ flat, const float* __restrict__ locations,
    const float* __restrict__ reg, const float* __restrict__ pemb,
    const float* __restrict__ pmar,
    float* __restrict__ embA, float* __restrict__ meta) {
    int img = blockIdx.x;
    int tid = threadIdx.x;
    const float* f = flat + (size_t)img * NFLAT;

    __shared__ float sV[1024];
    __shared__ int   sI[1024];
    __shared__ float tV[KTOP];
    __shared__ int   tI[KTOP];

    float prevV = 3.0e38f; int prevI = -1;   // +inf key: everything is "less"
    for (int r = 0; r < KTOP; ++r) {
        float bv = -3.0e38f; int bi = 0x7fffffff;
        for (int i = tid; i < NFLAT; i += 1024) {
            float v = f[i];
            // key (v desc, i asc) must be strictly less than previous winner
            bool lp = (v < prevV) || (v == prevV && i > prevI);
            if (lp && ((v > bv) || (v == bv && i < bi))) { bv = v; bi = i; }
        }
        sV[tid] = bv; sI[tid] = bi;
        __syncthreads();
        for (int s = 512; s > 0; s >>= 1) {
            if (tid < s) {
                float ov = sV[tid + s]; int oi = sI[tid + s];
                if (ov > sV[tid] || (ov == sV[tid] && oi < sI[tid])) { sV[tid] = ov; sI[tid] = oi; }
            }
            __syncthreads();
        }
        prevV = sV[0]; prevI = sI[0];
        if (tid == 0) { tV[r] = prevV; tI[r] = prevI; }
        __syncthreads();
    }

    // gather per-k data (rows 200..207 are zero padding)
    if (tid < KPAD) {
        float tv = (tid < KTOP) ? tV[tid] : -1.0f;
        int   ti = (tid < KTOP) ? tI[tid] : 0;
        int   li = ti / NC;                       // loc_idx
        int   g  = img * KPAD + tid;

        float lx = locations[li * 2 + 0], ly = locations[li * 2 + 1];
        const float* rg = reg + (size_t)img * 4 * HW;
        float x1 = fminf(fmaxf(lx - rg[0 * HW + li], 0.0f), (float)(IMGW - 1));
        float y1 = fminf(fmaxf(ly - rg[1 * HW + li], 0.0f), (float)(IMGH - 1));
        float x2 = fminf(fmaxf(lx + rg[2 * HW + li], 0.0f), (float)(IMGW - 1));
        float y2 = fminf(fmaxf(ly + rg[3 * HW + li], 0.0f), (float)(IMGH - 1));

        const float* pe = pemb + (size_t)img * EDIM * HW;
        float ss = 0.0f;
#pragma unroll
        for (int d = 0; d < EDIM; ++d) {
            float e = (tid < KTOP) ? pe[(size_t)d * HW + li] : 0.0f;
            embA[(size_t)g * EDIM + d] = e;
            ss += e * e;
        }
        meta[g * 8 + 0] = ss;
        meta[g * 8 + 1] = pmar[img * HW + li];
        meta[g * 8 + 2] = (tid < KTOP && tv > 0.0f) ? 1.0f : 0.0f;
        meta[g * 8 + 3] = 0.0f;
        meta[g * 8 + 4] = x1 * 0.25f;             // /MASK_STRIDE
        meta[g * 8 + 5] = y1 * 0.25f;
        meta[g * 8 + 6] = x2 * 0.25f;
        meta[g * 8 + 7] = y2 * 0.25f;
    }
}

// ---------------- WMMA GEMM (200x147456x32 f32) + exp/box/valid masking
// one wave per 16x16 output tile; 8 x v_wmma_f32_16x16x4_f32 chained over K=32.
// All 8 waves in a block share the same m-tile: A tile + metadata staged in LDS.
__global__ __launch_bounds__(256) void k_gemm_mask(
    const float* __restrict__ up, const float* __restrict__ psq,
    const float* __restrict__ embA, const float* __restrict__ meta,
    float* __restrict__ out) {
    __shared__ float sA[16 * EDIM];    // 512 floats: 16 rows x K=32
    __shared__ float sM[16 * 8];       // per-row metadata

    int t    = threadIdx.x;
    int wave = t >> 5, lane = t & 31;
    int nT    = blockIdx.x * 8 + wave;     // 0..9215 pixel tiles (exact)
    int img   = blockIdx.z;
    int mBase = blockIdx.y * 16;           // 0..192
    int row   = lane & 15, hi = lane >> 4;

    // stage shared A tile + metadata (once per block)
    {
        const float* Ag = embA + ((size_t)(img * KPAD + mBase)) * EDIM;
        sA[t]       = Ag[t];
        sA[t + 256] = Ag[t + 256];
        if (t < 128) {
            const float* Mg = meta + (size_t)(img * KPAD + mBase) * 8;
            sM[t] = Mg[t];
        }
    }
    __syncthreads();

    int p0 = nT * 16;
    const float* B = up + (size_t)img * EDIM * PP;   // [32][P]

    v8f acc = {0.f, 0.f, 0.f, 0.f, 0.f, 0.f, 0.f, 0.f};
#pragma unroll
    for (int kk = 0; kk < EDIM; kk += 4) {
        int ka = kk + 2 * hi;   // lanes0-15: K=kk,kk+1 ; lanes16-31: K=kk+2,kk+3
        v2f a, b;
        a.x = sA[row * EDIM + ka];
        a.y = sA[row * EDIM + ka + 1];
        b.x = B[(size_t)ka * PP + p0 + row];
        b.y = B[(size_t)(ka + 1) * PP + p0 + row];
        acc = __builtin_amdgcn_wmma_f32_16x16x4_f32(
            /*neg_a=*/false, a, /*neg_b=*/false, b,
            /*c_mod=*/(short)0, acc, /*reuse_a=*/false, /*reuse_b=*/false);
    }

    // pixel coordinates: tiles are aligned within an image row (384/16 == 24)
    int   ty = nT / NTILE_X;               // wave-uniform (scalar)
    int   tx = (nT - ty * NTILE_X) * 16;   // wave-uniform
    float fx = (float)(tx + row);
    float fy = (float)ty;
    int   p  = p0 + row;
    float ps = psq[img * PP + p];

    float* outp = out + (size_t)img * KTOP * PP + p;
#pragma unroll
    for (int r = 0; r < 8; ++r) {
        int kl = 8 * hi + r;               // local row 0..15 (C layout)
        int k  = mBase + kl;
        float eS = sM[kl * 8 + 0];
        float mg = sM[kl * 8 + 1];
        float vv = sM[kl * 8 + 2];
        float b0 = sM[kl * 8 + 4], b1 = sM[kl * 8 + 5];
        float b2 = sM[kl * 8 + 6], b3 = sM[kl * 8 + 7];

        float d2   = eS + ps - 2.0f * acc[r];
        float prob = __expf(-d2 * mg) * vv;
        bool  inb  = (fx >= b0) & (fx <= b2) & (fy >= b1) & (fy <= b3);
        float outv = inb ? prob : 0.0f;
        if (k < KTOP) outp[(size_t)k * PP] = outv;
    }
}

extern "C" void kernel_launch(void* const* d_in, const int* in_sizes, int n_in,
                              void* d_out, int out_size, void* d_ws, size_t ws_size,
                              hipStream_t stream) {
    (void)in_sizes; (void)n_in; (void)out_size; (void)ws_size;
    const float* locations = (const float*)d_in[0];   // [HW,2]
    const float* box_cls   = (const float*)d_in[1];   // [2,80,96,96]
    const float* box_reg   = (const float*)d_in[2];   // [2,4,96,96]
    const float* ctr       = (const float*)d_in[3];   // [2,1,96,96]
    const float* pemb      = (const float*)d_in[4];   // [2,32,96,96]
    const float* pmar      = (const float*)d_in[5];   // [2,1,96,96]
    const float* pix       = (const float*)d_in[6];   // [2,32,192,192]
    float* out = (float*)d_out;                       // [2,200,384,384]

    char* ws = (char*)d_ws;
    size_t off = 0;
    auto alloc = [&](size_t bytes) { size_t o = off; off += (bytes + 255) & ~(size_t)255; return o; };
    float* up   = (float*)(ws + alloc((size_t)NIMG * EDIM * PP * 4));   // 37.7 MB
    float* psq  = (float*)(ws + alloc((size_t)NIMG * PP * 4));          // 1.2 MB
    float* flat = (float*)(ws + alloc((size_t)NIMG * NFLAT * 4));       // 5.9 MB
    float* embA = (float*)(ws + alloc((size_t)NIMG * KPAD * EDIM * 4));
    float* meta = (float*)(ws + alloc((size_t)NIMG * KPAD * 8 * 4));

    k_upsample<<<(NIMG * EDIM * PP + 255) / 256, 256, 0, stream>>>(pix, up);
    k_psq<<<(NIMG * PP + 255) / 256, 256, 0, stream>>>(up, psq);
    k_scores<<<(NIMG * NFLAT + 255) / 256, 256, 0, stream>>>(box_cls, ctr, flat);
    k_topk<<<NIMG, 1024, 0, stream>>>(flat, locations, box_reg, pemb, pmar, embA, meta);
    dim3 g(PP / 16 / 8, KPAD / 16, NIMG);   // (1152, 13, 2), 8 waves/block
    k_gemm_mask<<<g, 256, 0, stream>>>(up, psq, embA, meta, out);
}